// GraphSAGE_42769284333575
// MI455X (gfx1250) — compile-verified
//
#include <hip/hip_runtime.h>

// GraphSAGE on MI455X (gfx1250, wave32).
// Bandwidth analysis: scatter phase moves ~410MB/layer through L2 atomics and
// dominates; the dense transform is only 1.64 GFLOP total, so we keep it in
// full fp32 precision using V_WMMA_F32_16X16X4_F32 (16x16 tile, K=4/step).

typedef __attribute__((ext_vector_type(2))) float v2f;
typedef __attribute__((ext_vector_type(8))) float v8f;

#define DD      64
#define TWO_D   128
#define ZPAD    132   // 128 + 4 -> conflict-free strided column reads
#define WPAD    132
#define OPAD    68    // 64 + 4

__global__ void sage_zero_kernel(float* __restrict__ p, int n) {
    int i = blockIdx.x * blockDim.x + threadIdx.x;
    if (i < n) p[i] = 0.0f;
}

__global__ void sage_deg_kernel(const int* __restrict__ dst, float* __restrict__ deg, int nE) {
    int e = blockIdx.x * blockDim.x + threadIdx.x;
    if (e < nE) atomicAdd(deg + dst[e], 1.0f);
}

__global__ void sage_invdeg_kernel(float* __restrict__ deg, int n) {
    int i = blockIdx.x * blockDim.x + threadIdx.x;
    if (i < n) {
        float d = deg[i];
        deg[i] = (d > 0.0f) ? (1.0f / d) : 0.0f;
    }
}

// thread t -> edge t/64, feature t%64: coalesced source-row read, f32 atomic add.
__global__ void sage_scatter_kernel(const float* __restrict__ h,
                                    const int* __restrict__ src,
                                    const int* __restrict__ dst,
                                    float* __restrict__ neigh, int nE) {
    int t = blockIdx.x * blockDim.x + threadIdx.x;
    int e = t >> 6;
    int d = t & 63;
    if (e < nE) {
        float v = h[(size_t)src[e] * DD + d];
        atomicAdd(neigh + (size_t)dst[e] * DD + d, v);
    }
}

// One workgroup = 128 threads = 4 waves; 16 nodes per workgroup.
// Wave w computes output columns [16w, 16w+16) as a 16x16 WMMA tile, K=128.
__global__ void __launch_bounds__(128)
sage_layer_kernel(const float* __restrict__ h,
                  const float* __restrict__ neigh,
                  const float* __restrict__ invdeg,
                  const float* __restrict__ Wk,    // [64][128] row-major
                  float* __restrict__ out, int nNodes) {
    __shared__ float sZ[16 * ZPAD];
    __shared__ float sW[64 * WPAD];
    __shared__ float sO[16 * OPAD];
    __shared__ float sInv[16];

    const int tid  = threadIdx.x;       // 0..127
    const int wave = tid >> 5;          // 0..3
    const int lane = tid & 31;
    const int half = lane >> 4;         // 0 or 1
    const int idx  = lane & 15;
    const int nodeBase = blockIdx.x * 16;

    // Stage weight slice 64x128 into LDS (coalesced).
    for (int i = tid; i < 64 * TWO_D; i += 128) {
        int r = i >> 7, c = i & 127;
        sW[r * WPAD + c] = Wk[i];
    }
    // Stage z = [h | neigh * inv_deg] (16 x 128), clamped for safety.
    for (int i = tid; i < 16 * DD; i += 128) {
        int r = i >> 6, c = i & 63;
        int node = nodeBase + r;
        if (node >= nNodes) node = nNodes - 1;
        float id = invdeg[node];
        sZ[r * ZPAD + c]      = h[(size_t)node * DD + c];
        sZ[r * ZPAD + 64 + c] = neigh[(size_t)node * DD + c] * id;
    }
    __syncthreads();

    // out[m][n] = sum_k z[m][k] * W[colBase+n][k]
    const int colBase = wave * 16;
    v8f acc = {};
#pragma unroll 4
    for (int kk = 0; kk < TWO_D; kk += 4) {
        const int k0 = kk + 2 * half;   // documented 16x4 f32 A layout
        v2f a, b;
        a.x = sZ[idx * ZPAD + k0];
        a.y = sZ[idx * ZPAD + k0 + 1];
        b.x = sW[(colBase + idx) * WPAD + k0];
        b.y = sW[(colBase + idx) * WPAD + k0 + 1];
        acc = __builtin_amdgcn_wmma_f32_16x16x4_f32(
            /*neg_a=*/false, a, /*neg_b=*/false, b,
            /*c_mod=*/(short)0, acc, /*reuse_a=*/false, /*reuse_b=*/false);
    }

    // C/D layout: VGPR r holds M=r (lanes 0-15) and M=r+8 (lanes 16-31), N=idx.
#pragma unroll
    for (int r = 0; r < 8; ++r) {
        int m = r + 8 * half;
        float v = acc[r];
        v = (v > 0.0f) ? v : 0.0f;                 // ReLU
        sO[m * OPAD + colBase + idx] = v;
    }
    __syncthreads();

    // Per-node L2 norm over the 64 outputs.
    if (tid < 16) {
        float s = 0.0f;
        for (int c = 0; c < DD; ++c) {
            float v = sO[tid * OPAD + c];
            s += v * v;
        }
        float nrm = sqrtf(s);
        sInv[tid] = 1.0f / fmaxf(nrm, 1e-12f);
    }
    __syncthreads();

    for (int i = tid; i < 16 * DD; i += 128) {
        int r = i >> 6, c = i & 63;
        int node = nodeBase + r;
        if (node < nNodes)
            out[(size_t)node * DD + c] = sO[r * OPAD + c] * sInv[r];
    }
}

extern "C" void kernel_launch(void* const* d_in, const int* in_sizes, int n_in,
                              void* d_out, int out_size, void* d_ws, size_t ws_size,
                              hipStream_t stream) {
    const float* x    = (const float*)d_in[0];   // [N, 64]
    const float* W    = (const float*)d_in[1];   // [2, 64, 128]
    const int*   esrc = (const int*)d_in[2];     // [E]
    const int*   edst = (const int*)d_in[3];     // [E]
    float*       out  = (float*)d_out;           // [N, 64]

    const int nNodes = in_sizes[0] / DD;         // 50000
    const int nEdges = in_sizes[2];              // 800000

    // Scratch layout: inv_deg[N] | neigh[N*64] | h1[N*64]  (~25.8 MB)
    float* invdeg = (float*)d_ws;
    float* neigh  = invdeg + nNodes;
    float* h1     = neigh + (size_t)nNodes * DD;

    const int B = 256;
    const int gN  = (nNodes + B - 1) / B;
    const int gE  = (nEdges + B - 1) / B;
    const int gND = ((nNodes * DD) + B - 1) / B;
    const int gED = ((nEdges * DD) + B - 1) / B;
    const int gTile = (nNodes + 15) / 16;

    // Degree -> inverse degree (deg accumulated in the invdeg buffer, then inverted).
    sage_zero_kernel<<<gN, B, 0, stream>>>(invdeg, nNodes);
    sage_deg_kernel<<<gE, B, 0, stream>>>(edst, invdeg, nEdges);
    sage_invdeg_kernel<<<gN, B, 0, stream>>>(invdeg, nNodes);

    const float* hin = x;
    float* houts[2] = { h1, out };
    for (int layer = 0; layer < 2; ++layer) {
        sage_zero_kernel<<<gND, B, 0, stream>>>(neigh, nNodes * DD);
        sage_scatter_kernel<<<gED, B, 0, stream>>>(hin, esrc, edst, neigh, nEdges);
        sage_layer_kernel<<<gTile, 128, 0, stream>>>(
            hin, neigh, invdeg, W + (size_t)layer * DD * TWO_D, houts[layer], nNodes);
        hin = houts[layer];
    }
}